// VectorQuantize_41704132444480
// MI455X (gfx1250) — compile-verified
//
#include <hip/hip_runtime.h>
#include <hip/hip_bf16.h>

// Problem constants (from reference)
#define BB   8
#define CC   256
#define HH   32
#define WW   32
#define HWN  (HH * WW)          // 1024
#define MPIX (BB * HH * WW)     // 8192 pixels
#define EE   256                // embedding dim
#define KK   8192               // codebook size
#define NOUT_ZQ (MPIX * EE)     // 2097152

typedef float v2f __attribute__((ext_vector_type(2)));
typedef float v4f __attribute__((ext_vector_type(4)));
typedef float v8f __attribute__((ext_vector_type(8)));
typedef int   v4i __attribute__((ext_vector_type(4)));

typedef __attribute__((address_space(1))) v4i* gptr_v4i;
typedef __attribute__((address_space(3))) v4i* lptr_v4i;

// ---------------------------------------------------------------------------
// Async copy helpers (CDNA5 ASYNCcnt path, with portable fallbacks)
// ---------------------------------------------------------------------------
__device__ __forceinline__ void async_copy_b128(const float* g, float* l) {
#if __has_builtin(__builtin_amdgcn_global_load_async_to_lds_b128)
  __builtin_amdgcn_global_load_async_to_lds_b128(
      (gptr_v4i)g, (lptr_v4i)l, 0, 0);
#else
  *(v4f*)l = *(const v4f*)g;
#endif
}

template <int N>
__device__ __forceinline__ void wait_asynccnt() {
#if __has_builtin(__builtin_amdgcn_s_wait_asynccnt)
  __builtin_amdgcn_s_wait_asynccnt(N);
#else
  asm volatile("s_wait_asynccnt %0" ::"i"(N) : "memory");
#endif
}

// Stage a contiguous 32-row (32 KB) embed tile into LDS: 256 threads x 8 x b128
__device__ __forceinline__ void stage_tile(const float* __restrict__ embed,
                                           int code0, float* dst, int t) {
#pragma unroll
  for (int i = 0; i < 8; ++i) {
    const int q = t + i * 256;                            // float4 index
    const float* g = embed + (size_t)code0 * EE + (size_t)q * 4;
    async_copy_b128(g, dst + q * 4);
  }
}

// ---------------------------------------------------------------------------
// Kernel 1: z_e[m, e] = sum_c z[b, c, hw] * proj_w[e, c] + proj_b[e]
// One 16x16 tile per wave, fp32 WMMA 16x16x4, K-loop = 256/4 = 64.
// A lane layout: lane l holds row m = (l&15), VGPR j holds K = (l>>4)*2 + j
// B lane layout: lane l holds col n = (l&15), VGPR j holds K = (l>>4)*2 + j
// D lane layout: lane l col n = (l&15), VGPR j row m = (l>>4)*8 + j
// ---------------------------------------------------------------------------
__global__ __launch_bounds__(256) void k_proj(const float* __restrict__ z,
                                              const float* __restrict__ pw,
                                              const float* __restrict__ pb,
                                              float* __restrict__ ze) {
  const int lane = threadIdx.x & 31;
  const int wave = threadIdx.x >> 5;
  const int tile = blockIdx.x * 8 + wave;   // 8192 tiles = 512 (M) x 16 (N)
  const int tm = tile >> 4;
  const int tn = tile & 15;
  const int n   = lane & 15;
  const int grp = lane >> 4;

  const int m  = tm * 16 + n;
  const int b  = m >> 10;
  const int hw = m & (HWN - 1);
  const float* zb = z + (size_t)b * CC * HWN + hw;   // + c*HWN per channel

  const int col = tn * 16 + n;
  const float bias = pb[col];

  v8f acc;
#pragma unroll
  for (int j = 0; j < 8; ++j) acc[j] = bias;

#pragma unroll 4
  for (int k = 0; k < 64; ++k) {
    const int c = k * 4 + grp * 2;
    v2f a;
    a[0] = zb[(size_t)(c + 0) * HWN];
    a[1] = zb[(size_t)(c + 1) * HWN];
    v2f bmat = *(const v2f*)(pw + (size_t)col * CC + c);
    acc = __builtin_amdgcn_wmma_f32_16x16x4_f32(false, a, false, bmat,
                                                (short)0, acc, false, false);
  }

  float* out = ze + (size_t)(tm * 16 + grp * 8) * EE + col;
#pragma unroll
  for (int j = 0; j < 8; ++j) out[(size_t)j * EE] = acc[j];
}

// ---------------------------------------------------------------------------
// Kernel 2: s[k] = ||embed[k]||^2   (one wave per codebook row)
// ---------------------------------------------------------------------------
__global__ __launch_bounds__(256) void k_norms(const float* __restrict__ embed,
                                               float* __restrict__ s) {
  const int lane = threadIdx.x & 31;
  const int row  = blockIdx.x * 8 + (threadIdx.x >> 5);
  const float* e = embed + (size_t)row * EE;
  float acc = 0.0f;
#pragma unroll
  for (int i = 0; i < EE / 32; ++i) {
    float v = e[lane + i * 32];
    acc += v * v;
  }
#pragma unroll
  for (int msk = 16; msk; msk >>= 1) acc += __shfl_xor(acc, msk, 32);
  if (lane == 0) s[row] = acc;
}

// ---------------------------------------------------------------------------
// Kernel 3: fused distance GEMM + argmin.
//   dist_var[m,k] = s[k] - 2 * ze[m,:] . embed[k,:]   (row-constant term dropped)
// Block = 8 waves = 128 pixel rows. A fragments live in registers (128 VGPRs
// per lane). B: 32-code embed tile staged in LDS via async-to-LDS, double
// buffered; each wave runs 2 N-tiles (acc0/acc1) per staged tile.
// ---------------------------------------------------------------------------
__global__ __launch_bounds__(256) void k_argmin(const float* __restrict__ ze,
                                                const float* __restrict__ embed,
                                                const float* __restrict__ s,
                                                int* __restrict__ ind) {
  __shared__ float btile[2][32 * EE];   // 2 x 32 KB
  const int t    = threadIdx.x;
  const int lane = t & 31;
  const int wave = t >> 5;
  const int n    = lane & 15;
  const int grp  = lane >> 4;
  const int mt   = blockIdx.x * 128 + wave * 16;

  // Preload this lane's A columns (c mod 4 in {grp*2, grp*2+1}) into registers.
  const float* arow = ze + (size_t)(mt + n) * EE + grp * 2;
  v2f areg[64];
#pragma unroll
  for (int k = 0; k < 64; ++k) areg[k] = *(const v2f*)(arow + k * 4);

  float bestv[8];
  int   besti[8];
#pragma unroll
  for (int j = 0; j < 8; ++j) { bestv[j] = 3.402823466e+38f; besti[j] = 0; }

  stage_tile(embed, 0, &btile[0][0], t);

  for (int ot = 0; ot < KK / 32; ++ot) {
    const int buf   = ot & 1;
    const int code0 = ot * 32;

    if (ot + 1 < KK / 32) {
      stage_tile(embed, code0 + 32, &btile[buf ^ 1][0], t);  // prefetch next
      wait_asynccnt<8>();   // previous 8 (current tile) complete, 8 in flight
    } else {
      wait_asynccnt<0>();
    }
    __syncthreads();        // current tile visible to all waves

    const float* bp = &btile[buf][n * EE] + grp * 2;
    v8f acc0 = {};
    v8f acc1 = {};
#pragma unroll
    for (int k = 0; k < 64; ++k) {
      v2f b0 = *(const v2f*)(bp + k * 4);             // codes code0 + n
      v2f b1 = *(const v2f*)(bp + 16 * EE + k * 4);   // codes code0 + 16 + n
      acc0 = __builtin_amdgcn_wmma_f32_16x16x4_f32(false, areg[k], false, b0,
                                                   (short)0, acc0, false, false);
      acc1 = __builtin_amdgcn_wmma_f32_16x16x4_f32(false, areg[k], false, b1,
                                                   (short)0, acc1, false, false);
    }
    __syncthreads();        // done reading buf before next prefetch overwrites

    const float sk0 = s[code0 + n];
    const float sk1 = s[code0 + 16 + n];
    const int   id0 = code0 + n;
    const int   id1 = code0 + 16 + n;
#pragma unroll
    for (int j = 0; j < 8; ++j) {
      const float d0 = sk0 - 2.0f * acc0[j];
      if (d0 < bestv[j]) { bestv[j] = d0; besti[j] = id0; }
      const float d1 = sk1 - 2.0f * acc1[j];
      if (d1 < bestv[j]) { bestv[j] = d1; besti[j] = id1; }
    }
  }

  // Min-reduce across the 16 N-lanes of each half-wave group (first-index tie)
#pragma unroll
  for (int j = 0; j < 8; ++j) {
    float v = bestv[j];
    int   i = besti[j];
#pragma unroll
    for (int msk = 1; msk <= 8; msk <<= 1) {
      const float ov = __shfl_xor(v, msk, 32);
      const int   oi = __shfl_xor(i, msk, 32);
      if (ov < v || (ov == v && oi < i)) { v = ov; i = oi; }
    }
    if (n == j) ind[mt + grp * 8 + j] = i;
  }
}

// ---------------------------------------------------------------------------
// Kernel 4: z_q gather, straight-through output, deterministic partial sums
// ---------------------------------------------------------------------------
__global__ __launch_bounds__(256) void k_gather(const float* __restrict__ ze,
                                                const float* __restrict__ embed,
                                                const int* __restrict__ ind,
                                                float* __restrict__ out_zq,
                                                float* __restrict__ partial) {
  __shared__ float red[256];
  const int row0 = blockIdx.x * 8;
  const int col  = threadIdx.x;
  float acc = 0.0f;
#pragma unroll
  for (int r = 0; r < 8; ++r) {
    const int row  = row0 + r;
    const int code = ind[row];
    const float zq  = embed[(size_t)code * EE + col];
    const float zev = ze[(size_t)row * EE + col];
    const float dq  = zq - zev;
    out_zq[(size_t)row * EE + col] = zev + dq;   // z_e + sg(z_q - z_e)
    acc += dq * dq;
  }
  red[threadIdx.x] = acc;
  __syncthreads();
  for (int st = 128; st; st >>= 1) {
    if (threadIdx.x < st) red[threadIdx.x] += red[threadIdx.x + st];
    __syncthreads();
  }
  if (threadIdx.x == 0) partial[blockIdx.x] = red[0];
}

// ---------------------------------------------------------------------------
// Kernel 5: final deterministic reduction -> diff = 2 * mean((zq-ze)^2)
// ---------------------------------------------------------------------------
__global__ __launch_bounds__(256) void k_diff(const float* __restrict__ partial,
                                              float* __restrict__ out) {
  __shared__ float red[256];
  float acc = 0.0f;
#pragma unroll
  for (int i = 0; i < 4; ++i) acc += partial[threadIdx.x + i * 256];
  red[threadIdx.x] = acc;
  __syncthreads();
  for (int st = 128; st; st >>= 1) {
    if (threadIdx.x < st) red[threadIdx.x] += red[threadIdx.x + st];
    __syncthreads();
  }
  if (threadIdx.x == 0)
    out[NOUT_ZQ] = 2.0f * red[0] / (float)NOUT_ZQ;   // (1 + COMMIT) * mean
}

// ---------------------------------------------------------------------------
// Kernel 6: indices -> output tail (exact integers as float values, K < 2^24)
// ---------------------------------------------------------------------------
__global__ __launch_bounds__(256) void k_idx(const int* __restrict__ ind,
                                             float* __restrict__ out) {
  const int i = blockIdx.x * 256 + threadIdx.x;
  if (i < MPIX) out[NOUT_ZQ + 1 + i] = (float)ind[i];
}

extern "C" void kernel_launch(void* const* d_in, const int* in_sizes, int n_in,
                              void* d_out, int out_size, void* d_ws, size_t ws_size,
                              hipStream_t stream) {
  const float* z      = (const float*)d_in[0];   // [B,C,H,W]
  const float* proj_w = (const float*)d_in[1];   // [E,C]
  const float* proj_b = (const float*)d_in[2];   // [E]
  const float* embed  = (const float*)d_in[3];   // [K,E]
  float* out = (float*)d_out;                    // [zq_st | diff | ind]

  char* ws = (char*)d_ws;
  float* ze      = (float*)ws;                             // 8,388,608 B
  float* snorm   = (float*)(ws + 8388608);                 // 32 KB
  int*   ind     = (int*)  (ws + 8388608 + 32768);         // 32 KB
  float* partial = (float*)(ws + 8388608 + 65536);         // 4 KB

  k_proj  <<<1024, 256, 0, stream>>>(z, proj_w, proj_b, ze);
  k_norms <<<1024, 256, 0, stream>>>(embed, snorm);
  k_argmin<<<  64, 256, 0, stream>>>(ze, embed, snorm, ind);
  k_gather<<<1024, 256, 0, stream>>>(ze, embed, ind, out, partial);
  k_diff  <<<   1, 256, 0, stream>>>(partial, out);
  k_idx   <<<  32, 256, 0, stream>>>(ind, out);
}